// warp_45999099740251
// MI455X (gfx1250) — compile-verified
//
#include <hip/hip_runtime.h>

// Bilinear warp (grid_sample, zeros padding, align_corners=False)
// frame: [B=16, C=3, H=720, W=1280] f32
// flow:  [B, 2, H, W] f32   (ch0 = x displacement, ch1 = y displacement)
// out:   [B, C, H, W] f32
//
// Memory-bound gather: ~472 MB mandatory traffic => ~20us floor at 23.3 TB/s.
// No matmul structure -> WMMA not applicable; optimize the wave32 memory path:
// b128 NT streams for flow/out, paired b64 gathers for the 4 corners.

typedef float v4f  __attribute__((ext_vector_type(4)));
typedef float v2fu __attribute__((ext_vector_type(2), aligned(4)));

constexpr int Bn = 16;
constexpr int Cn = 3;
constexpr int Hn = 720;
constexpr int Wn = 1280;
constexpr int HWn = Hn * Wn;

__global__ __launch_bounds__(Wn / 4) void warp_bilinear_kernel(
    const float* __restrict__ frame,
    const float* __restrict__ flow,
    float* __restrict__ out) {
  const int xg = threadIdx.x << 2;   // 0..1276, step 4 (one row per block)
  const int y  = blockIdx.x;         // 0..719
  const int b  = blockIdx.y;         // 0..15

  const size_t rowoff = (size_t)y * Wn + xg;

  // Flow is streamed exactly once: NT hint keeps WGP$/L2 capacity for the
  // frame gather working set (one image = 11 MB, fits easily in 192 MB L2).
  const float* flB = flow + (size_t)b * 2 * HWn;
  const v4f fx4 = __builtin_nontemporal_load((const v4f*)(flB + rowoff));
  const v4f fy4 = __builtin_nontemporal_load((const v4f*)(flB + HWn + rowoff));

  const float* frB = frame + (size_t)b * Cn * HWn;

  v4f o0, o1, o2;

  constexpr float kInvW = 2.0f / (float)(Wn - 1);
  constexpr float kInvH = 2.0f / (float)(Hn - 1);

#pragma unroll
  for (int j = 0; j < 4; ++j) {
    // --- reference math (normalize then grid_sample unnormalize) ---
    const float gx = (fx4[j] + (float)(xg + j)) * kInvW - 1.0f;
    const float gy = (fy4[j] + (float)y) * kInvH - 1.0f;
    const float ix = ((gx + 1.0f) * (float)Wn - 1.0f) * 0.5f;
    const float iy = ((gy + 1.0f) * (float)Hn - 1.0f) * 0.5f;

    const float x0f = floorf(ix);
    const float y0f = floorf(iy);
    const float wx1 = ix - x0f, wx0 = 1.0f - wx1;
    const float wy1 = iy - y0f, wy0 = 1.0f - wy1;

    const int x0 = (int)x0f, x1 = x0 + 1;
    const int y0 = (int)y0f, y1 = y0 + 1;

    const bool inx0 = (x0 >= 0) & (x0 < Wn);
    const bool inx1 = (x1 >= 0) & (x1 < Wn);
    const bool iny0 = (y0 >= 0) & (y0 < Hn);
    const bool iny1 = (y1 >= 0) & (y1 < Hn);

    // Clamped gather bases (always valid); zeros padding folded into weights.
    const int xb  = min(max(x0, 0), Wn - 2);
    const int yb0 = min(max(y0, 0), Hn - 1);
    const int yb1 = min(max(y1, 0), Hn - 1);
    const bool sel0 = (x0 == xb);  // whether pair.x corresponds to x0

    const float a00 = (inx0 & iny0) ? wx0 * wy0 : 0.0f;
    const float a01 = (inx1 & iny0) ? wx1 * wy0 : 0.0f;
    const float a10 = (inx0 & iny1) ? wx0 * wy1 : 0.0f;
    const float a11 = (inx1 & iny1) ? wx1 * wy1 : 0.0f;

    // Hoist the lane-routing select from the 12 loaded values onto the 4
    // weights (same boolean for every channel): each channel becomes a pure
    // 4-FMA dot against the raw loaded pairs.
    const float b00 = sel0 ? a00 : a01;
    const float b01 = sel0 ? a01 : a00;
    const float b10 = sel0 ? a10 : a11;
    const float b11 = sel0 ? a11 : a10;

    const size_t g0 = (size_t)yb0 * Wn + xb;
    const size_t g1 = (size_t)yb1 * Wn + xb;

    // Issue all 6 paired-corner gathers (b64) back-to-back so they clause.
    v2fu r0[3], r1[3];
#pragma unroll
    for (int c = 0; c < 3; ++c) {
      const float* p = frB + (size_t)c * HWn;
      r0[c] = *(const v2fu*)(p + g0);
      r1[c] = *(const v2fu*)(p + g1);
    }

#pragma unroll
    for (int c = 0; c < 3; ++c) {
      const float val = b00 * r0[c][0] + b01 * r0[c][1]
                      + b10 * r1[c][0] + b11 * r1[c][1];
      if (c == 0)      o0[j] = val;
      else if (c == 1) o1[j] = val;
      else             o2[j] = val;
    }
  }

  // Output written once: non-temporal b128 stores.
  float* outB = out + (size_t)b * Cn * HWn;
  __builtin_nontemporal_store(o0, (v4f*)(outB + rowoff));
  __builtin_nontemporal_store(o1, (v4f*)(outB + HWn + rowoff));
  __builtin_nontemporal_store(o2, (v4f*)(outB + 2 * (size_t)HWn + rowoff));
}

extern "C" void kernel_launch(void* const* d_in, const int* in_sizes, int n_in,
                              void* d_out, int out_size, void* d_ws, size_t ws_size,
                              hipStream_t stream) {
  const float* frame = (const float*)d_in[0];
  const float* flow  = (const float*)d_in[1];
  float* out = (float*)d_out;

  dim3 grid(Hn, Bn);        // (720 rows, 16 batches)
  dim3 block(Wn / 4);       // 320 threads = 10 wave32 per block, one row each
  warp_bilinear_kernel<<<grid, block, 0, stream>>>(frame, flow, out);
}